// RelationalAttLayer_63488206569617
// MI455X (gfx1250) — compile-verified
//
#include <hip/hip_runtime.h>
#include <hip/hip_bf16.h>

#define NN 50000
#define EE 800000
#define DD 128
#define HH 4
#define DHH 32

typedef __attribute__((ext_vector_type(2))) float v2f;
typedef __attribute__((ext_vector_type(8))) float v8f;

// ---------- order-preserving float<->uint encoding for atomic max ----------
__device__ __forceinline__ unsigned enc_f32(float f) {
    unsigned u = __float_as_uint(f);
    return (u & 0x80000000u) ? ~u : (u | 0x80000000u);
}
__device__ __forceinline__ float dec_f32(unsigned u) {
    return (u & 0x80000000u) ? __uint_as_float(u & 0x7FFFFFFFu)
                             : __uint_as_float(~u);
}

// ---------- GEMM: C[N,128] = A[N,128] @ B[128,128] (+ optional bias) -------
// One block = 16 output rows x 128 cols. 8 waves; wave w owns 16x16 tile at
// col w*16. A tile staged in LDS; fp32 WMMA 16x16x4, 32 K-steps.
__global__ __launch_bounds__(256) void gemm_wmma_kernel(
    const float* __restrict__ A, const float* __restrict__ B,
    const float* __restrict__ bias, float* __restrict__ C) {
    __shared__ float As[16 * 128];
    const int row0 = blockIdx.x * 16;
    const int tid  = threadIdx.x;
    const float* Ag = A + (size_t)row0 * DD;
#pragma unroll
    for (int i = 0; i < 8; ++i) As[tid + i * 256] = Ag[tid + i * 256];
    __syncthreads();

    const int wave = tid >> 5;
    const int lane = tid & 31;
    const int m    = lane & 15;          // row within tile (per half)
    const int half = lane >> 4;          // 0: K=0,1   1: K=2,3
    const int col  = wave * 16 + (lane & 15);

    v8f c = {};
#pragma unroll 8
    for (int kk = 0; kk < 32; ++kk) {
        const int k = kk * 4 + half * 2;
        v2f a, b;
        a.x = As[m * DD + k];
        a.y = As[m * DD + k + 1];
        b.x = B[(size_t)k * DD + col];
        b.y = B[(size_t)(k + 1) * DD + col];
        c = __builtin_amdgcn_wmma_f32_16x16x4_f32(
            /*neg_a=*/false, a, /*neg_b=*/false, b,
            /*c_mod=*/(short)0, c, /*reuse_a=*/false, /*reuse_b=*/false);
    }
    const float bv = bias ? bias[col] : 0.0f;
    float* Cg = C + (size_t)(row0 + half * 8) * DD + col;
#pragma unroll
    for (int j = 0; j < 8; ++j) Cg[(size_t)j * DD] = c[j] + bv;
}

// ---------- per-node attention scores: el/er [N,H] -------------------------
__global__ __launch_bounds__(256) void attn_scores_kernel(
    const float* __restrict__ feat, const float* __restrict__ al,
    const float* __restrict__ ar, float* __restrict__ el,
    float* __restrict__ er) {
    const int node = blockIdx.x * 8 + (threadIdx.x >> 5);
    const int lane = threadIdx.x & 31;
    if (node >= NN) return;
    const float* f = feat + (size_t)node * DD;
#pragma unroll
    for (int h = 0; h < HH; ++h) {
        const float v  = f[h * DHH + lane];
        float pl = v * al[h * DHH + lane];
        float pr = v * ar[h * DHH + lane];
#pragma unroll
        for (int off = 16; off > 0; off >>= 1) {
            pl += __shfl_xor(pl, off, 32);
            pr += __shfl_xor(pr, off, 32);
        }
        if (lane == 0) {
            el[node * HH + h] = pl;
            er[node * HH + h] = pr;
        }
    }
}

// ---------- init emax (encoded -inf == 0) and denom ------------------------
__global__ void init_kernel(unsigned* __restrict__ emax,
                            float* __restrict__ denom, int n) {
    const int i = blockIdx.x * blockDim.x + threadIdx.x;
    if (i < n) { emax[i] = 0u; denom[i] = 0.0f; }
}

// ---------- pass 1: e = leaky_relu(el[src]+er[dst]); segment max -----------
__global__ void edge_max_kernel(const int* __restrict__ src,
                                const int* __restrict__ dst,
                                const float* __restrict__ el,
                                const float* __restrict__ er,
                                float* __restrict__ ebuf,
                                unsigned* __restrict__ emax) {
    const int i = blockIdx.x * blockDim.x + threadIdx.x;
    if (i >= EE * HH) return;
    const int e = i >> 2, h = i & 3;
    const int s = src[e], d = dst[e];
    float v = el[s * HH + h] + er[d * HH + h];
    v = v > 0.0f ? v : 0.2f * v;                 // leaky relu, slope 0.2
    ebuf[i] = v;
    atomicMax(&emax[d * HH + h], enc_f32(v));
}

// ---------- pass 2: ex = exp(e - emax[dst]); segment sum (in-place) --------
__global__ void edge_expsum_kernel(const int* __restrict__ dst,
                                   float* __restrict__ ebuf,
                                   const unsigned* __restrict__ emax,
                                   float* __restrict__ denom) {
    const int i = blockIdx.x * blockDim.x + threadIdx.x;
    if (i >= EE * HH) return;
    const int e = i >> 2, h = i & 3;
    const int d = dst[e];
    const float ex = __expf(ebuf[i] - dec_f32(emax[d * HH + h]));
    ebuf[i] = ex;
    atomicAdd(&denom[d * HH + h], ex);
}

// ---------- pass 3: out[dst] += alpha * feat[src]  (one wave per edge) -----
__global__ __launch_bounds__(256) void edge_scatter_kernel(
    const int* __restrict__ src, const int* __restrict__ dst,
    const float* __restrict__ ebuf, const float* __restrict__ denom,
    const float* __restrict__ feat, float* __restrict__ out) {
    const int e = blockIdx.x * 8 + (threadIdx.x >> 5);
    if (e >= EE) return;
    const int lane = threadIdx.x & 31;
    const int s = src[e], d = dst[e];
    const float* fs = feat + (size_t)s * DD;
    float* od = out + (size_t)d * DD;
#pragma unroll
    for (int h = 0; h < HH; ++h) {
        const float alpha = ebuf[e * HH + h] / denom[d * HH + h];
        atomicAdd(&od[h * DHH + lane], alpha * fs[h * DHH + lane]);
    }
}

// ---------- final ReLU ------------------------------------------------------
__global__ void relu_kernel(float* __restrict__ out, int n) {
    const int i = blockIdx.x * blockDim.x + threadIdx.x;
    if (i < n) out[i] = fmaxf(out[i], 0.0f);
}

extern "C" void kernel_launch(void* const* d_in, const int* in_sizes, int n_in,
                              void* d_out, int out_size, void* d_ws, size_t ws_size,
                              hipStream_t stream) {
    const float* x    = (const float*)d_in[0];
    const float* W0   = (const float*)d_in[1];
    const float* al0  = (const float*)d_in[2];
    const float* ar0  = (const float*)d_in[3];
    const float* W1   = (const float*)d_in[4];
    const float* al1  = (const float*)d_in[5];
    const float* ar1  = (const float*)d_in[6];
    const float* Wl   = (const float*)d_in[7];
    const float* bias = (const float*)d_in[8];
    const int* src[2] = {(const int*)d_in[9],  (const int*)d_in[11]};
    const int* dst[2] = {(const int*)d_in[10], (const int*)d_in[12]};
    const float* Wr[2] = {W0, W1};
    const float* al[2] = {al0, al1};
    const float* ar[2] = {ar0, ar1};

    float* out = (float*)d_out;
    char*  ws  = (char*)d_ws;
    // workspace layout (bytes)
    float*    feat  = (float*)(ws);                       // 50000*128*4 = 25,600,000
    float*    el    = (float*)(ws + 25600000);            // 200000*4   =    800,000
    float*    er    = (float*)(ws + 26400000);            //               800,000
    unsigned* emax  = (unsigned*)(ws + 27200000);         //               800,000
    float*    denom = (float*)(ws + 28000000);            //               800,000
    float*    ebuf  = (float*)(ws + 28800000);            // 3.2M*4     = 12,800,000
    // total = 41,600,000 bytes

    // out = x @ loop_weight + bias
    gemm_wmma_kernel<<<NN / 16, 256, 0, stream>>>(x, Wl, bias, out);

    for (int r = 0; r < 2; ++r) {
        gemm_wmma_kernel<<<NN / 16, 256, 0, stream>>>(x, Wr[r], nullptr, feat);
        attn_scores_kernel<<<NN / 8, 256, 0, stream>>>(feat, al[r], ar[r], el, er);
        init_kernel<<<(NN * HH + 255) / 256, 256, 0, stream>>>(emax, denom, NN * HH);
        edge_max_kernel<<<(EE * HH + 255) / 256, 256, 0, stream>>>(
            src[r], dst[r], el, er, ebuf, emax);
        edge_expsum_kernel<<<(EE * HH + 255) / 256, 256, 0, stream>>>(
            dst[r], ebuf, emax, denom);
        edge_scatter_kernel<<<EE / 8, 256, 0, stream>>>(
            src[r], dst[r], ebuf, denom, feat, out);
    }

    relu_kernel<<<(NN * DD + 255) / 256, 256, 0, stream>>>(out, NN * DD);
}